// FeaturesEncoder_dualPatches_untie_39822936768828
// MI455X (gfx1250) — compile-verified
//
#include <hip/hip_runtime.h>
#include <math.h>

// Problem constants (match reference)
#define B_  16
#define S_  512
#define D_  512
#define H_  8
#define L_  6
#define DF_ 2048
#define DK_ 64

typedef float v2f __attribute__((ext_vector_type(2)));
typedef float v8f __attribute__((ext_vector_type(8)));

static __device__ __forceinline__ v8f wmma4(v2f a, v2f b, v8f c) {
  // D = A(16x4 f32) * B(4x16 f32) + C(16x16 f32), full fp32 precision
  return __builtin_amdgcn_wmma_f32_16x16x4_f32(
      /*neg_a=*/false, a, /*neg_b=*/false, b,
      /*c_mod=*/(short)0, c, /*reuse_a=*/false, /*reuse_b=*/false);
}

static __device__ __forceinline__ v8f zero8() {
  v8f z;
#pragma unroll
  for (int e = 0; e < 8; ++e) z[e] = 0.0f;
  return z;
}

// ---------------------------------------------------------------------------
// Generic fp32 WMMA GEMM:  C[z] = act( alpha * A[z]@B[z] + bias + resid )
// Block tile: 128(M) x 128(N), BK=16, 256 threads = 8 waves (4x2 wave grid).
// Each wave owns a 32x64 slab = 2 m-tiles x 4 n-tiles = 8 WMMA accumulators.
// B staged pair-interleaved along K: tBp[k/2][n][2] -> one aligned ds_load_b64
// per B fragment (no packing movs). A rows padded to 18 so v2f loads are
// 8B-aligned and bank-spread.
// All shapes used here divide tiles exactly (M%128==0, N%128==0, K%16==0).
// act: 0 = none, 1 = exact GELU.
// ---------------------------------------------------------------------------
__global__ __launch_bounds__(256) void gemm_f32_wmma(
    const float* __restrict__ A, const float* __restrict__ Bm,
    const float* __restrict__ bias, const float* __restrict__ resid,
    float* __restrict__ C,
    int M, int N, int K, int lda, int ldb, int ldc,
    long long strideA, long long strideB, long long strideC,
    int transB, int act, float alpha)
{
  __shared__ float tA[128][18];      // row stride 72B: 8B-aligned, bank-spread
  __shared__ float tBp[8][128][2];   // [k/2][n][k&1] pair-interleaved

  const int z = blockIdx.z;
  A  += (long long)z * strideA;
  Bm += (long long)z * strideB;
  C  += (long long)z * strideC;

  const int n0   = blockIdx.x * 128;
  const int m0   = blockIdx.y * 128;
  const int tid  = threadIdx.x;
  const int lane = tid & 31;
  const int wave = tid >> 5;
  const int hlf  = lane >> 4;          // which 16-lane half
  const int l15  = lane & 15;
  const int wm   = (wave >> 1) * 32;   // wave row base within tile (2 m-tiles)
  const int wn   = (wave & 1) * 64;    // wave col base within tile (4 n-tiles)

  v8f acc[2][4];
#pragma unroll
  for (int mt = 0; mt < 2; ++mt)
#pragma unroll
    for (int t = 0; t < 4; ++t) acc[mt][t] = zero8();

  const int ksteps = K >> 4;
  for (int kt = 0; kt < ksteps; ++kt) {
    const int k0 = kt << 4;
    // Stage A tile 128x16 (2048 elems / 256 threads)
#pragma unroll
    for (int i = 0; i < 8; ++i) {
      int idx = tid + i * 256;
      int r = idx >> 4, c = idx & 15;
      tA[r][c] = A[(m0 + r) * lda + (k0 + c)];
    }
    // Stage B tile 16x128, pair-interleaved along K (2048 elems / 256 threads)
    if (!transB) {
#pragma unroll
      for (int i = 0; i < 8; ++i) {
        int idx = tid + i * 256;
        int r = idx >> 7, c = idx & 127;
        tBp[r >> 1][c][r & 1] = Bm[(k0 + r) * ldb + (n0 + c)];
      }
    } else {
#pragma unroll
      for (int i = 0; i < 8; ++i) {
        int idx = tid + i * 256;
        int r = idx >> 7, c = idx & 127;
        tBp[r >> 1][c][r & 1] = Bm[(n0 + c) * ldb + (k0 + r)];
      }
    }
    __syncthreads();

    // Pull next K tile toward the near caches while we compute this one.
    if (kt + 1 < ksteps) {
      __builtin_prefetch(&A[(m0 + (tid >> 1)) * lda + k0 + 16 + 8 * (tid & 1)], 0, 3);
      __builtin_prefetch(&Bm[transB ? ((n0 + (tid & 127)) * ldb + k0 + 16)
                                    : ((k0 + 16 + (tid & 15)) * ldb + n0 + 8 * (tid >> 4))], 0, 3);
    }

#pragma unroll
    for (int kk = 0; kk < 16; kk += 4) {
      const int kp = (kk >> 1) + hlf;     // pair index; 2*kp = kk + 2*hlf (even)
      // Batch all fragment loads for this k-step, then issue 8 WMMAs.
      v2f a0 = *(const v2f*)&tA[wm + l15][2 * kp];
      v2f a1 = *(const v2f*)&tA[wm + 16 + l15][2 * kp];
      v2f bf[4];
#pragma unroll
      for (int t = 0; t < 4; ++t)
        bf[t] = *(const v2f*)&tBp[kp][wn + t * 16 + l15][0];
#pragma unroll
      for (int t = 0; t < 4; ++t) acc[0][t] = wmma4(a0, bf[t], acc[0][t]);
#pragma unroll
      for (int t = 0; t < 4; ++t) acc[1][t] = wmma4(a1, bf[t], acc[1][t]);
    }
    __syncthreads();
  }

  // Epilogue: D-fragment layout -> lane (l15 + 16*hlf) holds rows i+8*hlf.
#pragma unroll
  for (int mt = 0; mt < 2; ++mt) {
#pragma unroll
    for (int t = 0; t < 4; ++t) {
      const int c = n0 + wn + t * 16 + l15;
      const float bv = bias ? bias[c] : 0.0f;
#pragma unroll
      for (int i = 0; i < 8; ++i) {
        const int r = m0 + wm + mt * 16 + i + 8 * hlf;
        float v = alpha * acc[mt][t][i] + bv;
        if (resid) v += resid[r * ldc + c];
        if (act == 1) v = 0.5f * v * (1.0f + erff(v * 0.70710678118654752f));
        C[r * ldc + c] = v;
      }
    }
  }
}

// ---------------------------------------------------------------------------
// Fused flash attention with PE bias, fp32 WMMA.
// Grid: (S/64, H, B). Block: 128 threads = 4 waves; each wave owns 16 q rows.
// Q/K/V/O layout: [B, S, D] with head slice at column head*DK.
// K^T and V tiles staged pair-interleaved along the contraction dim so each
// B fragment is one aligned ds_load_b64.
// ---------------------------------------------------------------------------
__global__ __launch_bounds__(128) void flash_attn(
    const float* __restrict__ Qg, const float* __restrict__ Kg,
    const float* __restrict__ Vg, const float* __restrict__ PE,
    float* __restrict__ Og, float inv_scale)
{
  __shared__ float sQ[64][66];       // [q][d], stride 264B: aligned + spread
  __shared__ float sKtP[32][65][2];  // [d/2][key][d&1]
  __shared__ float sVP[32][65][2];   // [key/2][d][key&1]
  __shared__ float sP[4][16][64];    // wave-private P staging (A operand)

  const int qb0  = blockIdx.x * 64;
  const int head = blockIdx.y;
  const int b    = blockIdx.z;
  const int base = b * (S_ * D_) + head * DK_;
  const float* Qp = Qg + base;
  const float* Kp = Kg + base;
  const float* Vp = Vg + base;
  float*       Op = Og + base;
  const float* PEp = PE + head * (S_ * S_);

  const int tid  = threadIdx.x;
  const int lane = tid & 31;
  const int wave = tid >> 5;
  const int hlf  = lane >> 4;
  const int l15  = lane & 15;

  // Stage the 64x64 Q block once.
#pragma unroll
  for (int i = 0; i < 32; ++i) {
    int idx = tid + i * 128;
    int r = idx >> 6, c = idx & 63;
    sQ[r][c] = Qp[(qb0 + r) * D_ + c];
  }

  v8f o[4];
#pragma unroll
  for (int t = 0; t < 4; ++t) o[t] = zero8();
  float mrun[8], lrun[8];
#pragma unroll
  for (int i = 0; i < 8; ++i) { mrun[i] = -INFINITY; lrun[i] = 0.0f; }

  for (int kb = 0; kb < S_ / 64; ++kb) {
    __syncthreads();  // previous chunk fully consumed (also covers Q staging)
#pragma unroll
    for (int i = 0; i < 32; ++i) {
      int idx = tid + i * 128;
      int r = idx >> 6, c = idx & 63;   // r = key, c = d  (coalesced reads)
      sKtP[c >> 1][r][c & 1] = Kp[(kb * 64 + r) * D_ + c];
      sVP[r >> 1][c][r & 1]  = Vp[(kb * 64 + r) * D_ + c];
    }
    __syncthreads();

    // S = Q * K^T  (contract over d = 64)
    v8f s[4];
#pragma unroll
    for (int t = 0; t < 4; ++t) s[t] = zero8();
#pragma unroll
    for (int kk = 0; kk < 64; kk += 4) {
      const int kp = (kk >> 1) + hlf;
      v2f af; af.x = sQ[wave * 16 + l15][2 * kp]; af.y = sQ[wave * 16 + l15][2 * kp + 1];
      v2f bf[4];
#pragma unroll
      for (int t = 0; t < 4; ++t)
        bf[t] = *(const v2f*)&sKtP[kp][t * 16 + l15][0];
#pragma unroll
      for (int t = 0; t < 4; ++t) s[t] = wmma4(af, bf[t], s[t]);
    }

    // scale + PE bias (D-fragment addressing)
#pragma unroll
    for (int t = 0; t < 4; ++t) {
      const int col = kb * 64 + t * 16 + l15;
#pragma unroll
      for (int i = 0; i < 8; ++i) {
        const int row = qb0 + wave * 16 + i + 8 * hlf;
        s[t][i] = s[t][i] * inv_scale + PEp[row * S_ + col];
      }
    }

    // Online softmax. Row i+8*hlf lives across 16 lanes of one half-wave:
    // xor masks 1..8 never cross the half boundary -> valid row reduction.
    float alph[8];
#pragma unroll
    for (int i = 0; i < 8; ++i) {
      float v = fmaxf(fmaxf(s[0][i], s[1][i]), fmaxf(s[2][i], s[3][i]));
      v = fmaxf(v, __shfl_xor(v, 1, 32));
      v = fmaxf(v, __shfl_xor(v, 2, 32));
      v = fmaxf(v, __shfl_xor(v, 4, 32));
      v = fmaxf(v, __shfl_xor(v, 8, 32));
      const float mnew = fmaxf(mrun[i], v);
      alph[i] = __expf(mrun[i] - mnew);
      mrun[i] = mnew;
    }
#pragma unroll
    for (int i = 0; i < 8; ++i) {
      float acc = 0.0f;
#pragma unroll
      for (int t = 0; t < 4; ++t) {
        const float p = __expf(s[t][i] - mrun[i]);
        s[t][i] = p;
        acc += p;
      }
      acc += __shfl_xor(acc, 1, 32);
      acc += __shfl_xor(acc, 2, 32);
      acc += __shfl_xor(acc, 4, 32);
      acc += __shfl_xor(acc, 8, 32);
      lrun[i] = lrun[i] * alph[i] + acc;
#pragma unroll
      for (int t = 0; t < 4; ++t) o[t][i] *= alph[i];
    }

    // Re-fragment P (D-layout -> A-layout) through wave-private LDS.
#pragma unroll
    for (int t = 0; t < 4; ++t)
#pragma unroll
      for (int i = 0; i < 8; ++i)
        sP[wave][i + 8 * hlf][t * 16 + l15] = s[t][i];
    // Same-wave LDS ops are in order: store->load is safe without a barrier.

    // O += P * V  (contract over key = 64)
#pragma unroll
    for (int kk = 0; kk < 64; kk += 4) {
      const int kp = (kk >> 1) + hlf;
      v2f af = *(const v2f*)&sP[wave][l15][2 * kp];   // row stride 256B: aligned
      v2f bf[4];
#pragma unroll
      for (int t = 0; t < 4; ++t)
        bf[t] = *(const v2f*)&sVP[kp][t * 16 + l15][0];
#pragma unroll
      for (int t = 0; t < 4; ++t) o[t] = wmma4(af, bf[t], o[t]);
    }
  }

  // Normalize and write out (head-interleaved [B,S,D] layout).
#pragma unroll
  for (int i = 0; i < 8; ++i) {
    const float inv = 1.0f / lrun[i];
    const int row = qb0 + wave * 16 + i + 8 * hlf;
#pragma unroll
    for (int t = 0; t < 4; ++t)
      Op[row * D_ + t * 16 + l15] = o[t][i] * inv;
  }
}

// ---------------------------------------------------------------------------
// LayerNorm: 1 wave per row (D=512 -> 16 elems/lane), biased variance.
// Block = 256 threads = 8 rows.
// ---------------------------------------------------------------------------
__global__ __launch_bounds__(256) void layernorm_rows(
    float* __restrict__ X, const float* __restrict__ g, const float* __restrict__ bb)
{
  const int tid  = threadIdx.x;
  const int lane = tid & 31;
  const int wave = tid >> 5;
  const int row  = blockIdx.x * 8 + wave;
  float* xp = X + row * D_;

  float v[16];
  float s = 0.0f, s2 = 0.0f;
#pragma unroll
  for (int j = 0; j < 16; ++j) {
    const float x = xp[lane + j * 32];
    v[j] = x; s += x; s2 += x * x;
  }
#pragma unroll
  for (int off = 1; off < 32; off <<= 1) {
    s  += __shfl_xor(s,  off, 32);
    s2 += __shfl_xor(s2, off, 32);
  }
  const float mean = s * (1.0f / D_);
  const float var  = s2 * (1.0f / D_) - mean * mean;
  const float rstd = rsqrtf(var + 1e-6f);
#pragma unroll
  for (int j = 0; j < 16; ++j) {
    const int c = lane + j * 32;
    xp[c] = (v[j] - mean) * rstd * g[c] + bb[c];
  }
}

// ---------------------------------------------------------------------------
// h = x * sqrt(D); pe_out[b,s,d] = pe_table[s,d]  (broadcast half of d_out)
// ---------------------------------------------------------------------------
__global__ void init_h_and_peout(const float* __restrict__ x,
                                 const float* __restrict__ pe,
                                 float* __restrict__ h,
                                 float* __restrict__ pe_out)
{
  const long long i = (long long)blockIdx.x * blockDim.x + threadIdx.x;
  if (i < (long long)B_ * S_ * D_) {
    h[i] = x[i] * 22.62741699796952f;          // sqrt(512)
    pe_out[i] = pe[i & (S_ * D_ - 1)];         // S*D is a power of two
  }
}

extern "C" void kernel_launch(void* const* d_in, const int* in_sizes, int n_in,
                              void* d_out, int out_size, void* d_ws, size_t ws_size,
                              hipStream_t stream) {
  const float* x      = (const float*)d_in[0];
  const float* pe     = (const float*)d_in[1];
  const float* pe_q_w = (const float*)d_in[2];
  const float* pe_q_b = (const float*)d_in[3];
  const float* pe_k_w = (const float*)d_in[4];
  const float* pe_k_b = (const float*)d_in[5];
  const float* Wq = (const float*)d_in[6];   const float* bq = (const float*)d_in[7];
  const float* Wk = (const float*)d_in[8];   const float* bk = (const float*)d_in[9];
  const float* Wv = (const float*)d_in[10];  const float* bv = (const float*)d_in[11];
  const float* Wo = (const float*)d_in[12];  const float* bo = (const float*)d_in[13];
  const float* ln1g = (const float*)d_in[14]; const float* ln1b = (const float*)d_in[15];
  const float* W1 = (const float*)d_in[16];  const float* b1 = (const float*)d_in[17];
  const float* W2 = (const float*)d_in[18];  const float* b2 = (const float*)d_in[19];
  const float* ln2g = (const float*)d_in[20]; const float* ln2b = (const float*)d_in[21];

  float* out = (float*)d_out;
  float* ws  = (float*)d_ws;

  const long long SD  = (long long)S_ * D_;
  const long long BSD = (long long)B_ * S_ * D_;

  // Workspace layout (floats): ~128 MB total
  float* pq   = ws;                                  // S*D
  float* pk   = pq + SD;                             // S*D
  float* spe  = pk + SD;                             // H*S*S
  float* hbuf = spe + (long long)H_ * S_ * S_;       // B*S*D
  float* h1   = hbuf + BSD;                          // B*S*D
  float* obuf = h1 + BSD;                            // B*S*D
  float* tmp  = obuf + BSD;                          // max(3*BSD, B*S*DF)
  float* qb_  = tmp;
  float* kb_  = tmp + BSD;
  float* vb_  = tmp + 2 * BSD;
  float* f1   = tmp;                                 // reuses q/k/v region

  const float inv_scale = 1.0f / sqrtf(2.0f * DK_);  // 1/sqrt(128)
  const dim3 blk256(256);
  const int MBS = B_ * S_;                           // 8192 rows

  // 1) h = x*sqrt(D); pe_out broadcast
  init_h_and_peout<<<dim3((unsigned)((BSD + 255) / 256)), blk256, 0, stream>>>(
      x, pe, hbuf, out + BSD);

  // 2) pq = pe@pe_q_w + b ; pk = pe@pe_k_w + b     (512x512x512)
  gemm_f32_wmma<<<dim3(D_ / 128, S_ / 128, 1), blk256, 0, stream>>>(
      pe, pe_q_w, pe_q_b, nullptr, pq, S_, D_, D_, D_, D_, D_, 0, 0, 0, 0, 0, 1.0f);
  gemm_f32_wmma<<<dim3(D_ / 128, S_ / 128, 1), blk256, 0, stream>>>(
      pe, pe_k_w, pe_k_b, nullptr, pk, S_, D_, D_, D_, D_, D_, 0, 0, 0, 0, 0, 1.0f);

  // 3) scores_pe[h] = (pq_h @ pk_h^T) / scale      (z-batched over heads, transB)
  gemm_f32_wmma<<<dim3(S_ / 128, S_ / 128, H_), blk256, 0, stream>>>(
      pq, pk, nullptr, nullptr, spe, S_, S_, DK_, D_, D_, S_,
      (long long)DK_, (long long)DK_, (long long)S_ * S_, 1, 0, inv_scale);

  // 4) transformer layers
  for (int l = 0; l < L_; ++l) {
    const float* wq = Wq + (long long)l * D_ * D_;  const float* bql = bq + l * D_;
    const float* wk = Wk + (long long)l * D_ * D_;  const float* bkl = bk + l * D_;
    const float* wv = Wv + (long long)l * D_ * D_;  const float* bvl = bv + l * D_;
    const float* wo = Wo + (long long)l * D_ * D_;  const float* bol = bo + l * D_;
    const float* w1 = W1 + (long long)l * D_ * DF_; const float* b1l = b1 + l * DF_;
    const float* w2 = W2 + (long long)l * DF_ * D_; const float* b2l = b2 + l * D_;

    // QKV projections
    gemm_f32_wmma<<<dim3(D_ / 128, MBS / 128, 1), blk256, 0, stream>>>(
        hbuf, wq, bql, nullptr, qb_, MBS, D_, D_, D_, D_, D_, 0, 0, 0, 0, 0, 1.0f);
    gemm_f32_wmma<<<dim3(D_ / 128, MBS / 128, 1), blk256, 0, stream>>>(
        hbuf, wk, bkl, nullptr, kb_, MBS, D_, D_, D_, D_, D_, 0, 0, 0, 0, 0, 1.0f);
    gemm_f32_wmma<<<dim3(D_ / 128, MBS / 128, 1), blk256, 0, stream>>>(
        hbuf, wv, bvl, nullptr, vb_, MBS, D_, D_, D_, D_, D_, 0, 0, 0, 0, 0, 1.0f);

    // fused attention with PE bias
    flash_attn<<<dim3(S_ / 64, H_, B_), dim3(128), 0, stream>>>(
        qb_, kb_, vb_, spe, obuf, inv_scale);

    // o @ Wo + bo + h  -> h1 (pre-LN), then LN1 in place
    gemm_f32_wmma<<<dim3(D_ / 128, MBS / 128, 1), blk256, 0, stream>>>(
        obuf, wo, bol, hbuf, h1, MBS, D_, D_, D_, D_, D_, 0, 0, 0, 0, 0, 1.0f);
    layernorm_rows<<<dim3(MBS / 8), blk256, 0, stream>>>(h1, ln1g + l * D_, ln1b + l * D_);

    // FFN: f1 = gelu(h1@W1 + b1) ; h = LN2(f1@W2 + b2 + h1)
    gemm_f32_wmma<<<dim3(DF_ / 128, MBS / 128, 1), blk256, 0, stream>>>(
        h1, w1, b1l, nullptr, f1, MBS, DF_, D_, D_, DF_, DF_, 0, 0, 1.0f ? 0 : 0, 0, 1, 1.0f);
    gemm_f32_wmma<<<dim3(D_ / 128, MBS / 128, 1), blk256, 0, stream>>>(
        f1, w2, b2l, h1, hbuf, MBS, D_, DF_, DF_, D_, D_, 0, 0, 0, 0, 0, 1.0f);
    layernorm_rows<<<dim3(MBS / 8), blk256, 0, stream>>>(hbuf, ln2g + l * D_, ln2b + l * D_);
  }

  // 5) first half of d_out = final h
  hipMemcpyAsync(out, hbuf, BSD * sizeof(float), hipMemcpyDeviceToDevice, stream);
}